// GreedyInference_31619549233640
// MI455X (gfx1250) — compile-verified
//
#include <hip/hip_runtime.h>
#include <stdint.h>

typedef float v2f __attribute__((ext_vector_type(2)));
typedef float v8f __attribute__((ext_vector_type(8)));

#define HDIM 1024   // H == D == J == 1024
#define BDIM 8      // batch
#define MAXSYM 4

__device__ __forceinline__ v8f wmma_f32(v2f a, v2f b, v8f c) {
  // D = A(16x4) * B(4x16) + C, fp32 -> v_wmma_f32_16x16x4_f32
  return __builtin_amdgcn_wmma_f32_16x16x4_f32(false, a, false, b, (short)0, c, false, false);
}

__device__ __forceinline__ unsigned long long packKey(float v, int idx) {
  unsigned u = __float_as_uint(v);
  u = (u & 0x80000000u) ? ~u : (u | 0x80000000u);            // orderable float
  return ((unsigned long long)u << 32) |
         (unsigned long long)(0xFFFFFFFFu - (unsigned)idx);   // ties -> smallest idx
}

__device__ __forceinline__ float sigmoidf_(float x) { return 1.0f / (1.0f + expf(-x)); }

// ---------------- Phase 1: enc_proj[m, :] = enc[b,:,t] @ W_enc + b_joint ----------------
// enc layout (B, D, T): enc[b*D*T + d*T + t];  m = b*T + t.  Rows clamped, stores guarded.
__global__ void enc_proj_kernel(const float* __restrict__ enc,
                                const float* __restrict__ W_enc,
                                const float* __restrict__ b_joint,
                                float* __restrict__ encp, int T, int M) {
  int wave = threadIdx.x >> 5, lane = threadIdx.x & 31;
  int n = lane & 15;
  int col = blockIdx.x * 128 + wave * 32 + n;     // tiles: col, col+16
  int m0 = blockIdx.y * 16;
  int kk = (lane < 16) ? 0 : 2;
  int mA = m0 + (lane & 15);
  if (mA >= M) mA = M - 1;                        // clamp: dup rows discarded at store
  int bA = mA / T, tA = mA % T;
  const float* aBase = enc + (size_t)bA * HDIM * T + tA;

  v8f acc0 = {}, acc1 = {};
  int aoff = kk * T;
  int woff = kk * HDIM + col;
#pragma unroll 8
  for (int k0 = 0; k0 < HDIM; k0 += 4) {
    v2f a, b0, b1;
    a.x = aBase[aoff];
    a.y = aBase[aoff + T];
    b0.x = W_enc[woff];       b0.y = W_enc[woff + HDIM];
    b1.x = W_enc[woff + 16];  b1.y = W_enc[woff + HDIM + 16];
    acc0 = wmma_f32(a, b0, acc0);
    acc1 = wmma_f32(a, b1, acc1);
    aoff += 4 * T;
    woff += 4 * HDIM;
  }
  float bj0 = b_joint[col], bj1 = b_joint[col + 16];
  int rbase = (lane < 16) ? 0 : 8;
#pragma unroll
  for (int r = 0; r < 8; ++r) {
    int m = m0 + rbase + r;
    if (m < M) {
      encp[(size_t)m * HDIM + col] = acc0[r] + bj0;
      encp[(size_t)m * HDIM + col + 16] = acc1[r] + bj1;
    }
  }
}

// ---------------- Per-step kernel 1: fused LSTM (z = x@Wx + h@Wh + b, pointwise) ----------
// grid = H/32 blocks, 128 threads (4 waves; wave w = quadrant w of i/f/g/o, 32 cols each)
__global__ void lstm_kernel(const float* __restrict__ wsx, const float* __restrict__ wsh,
                            const float* __restrict__ Wx, const float* __restrict__ Wh,
                            const float* __restrict__ bias, const float* __restrict__ wsc,
                            float* __restrict__ wsh2, float* __restrict__ wsc2) {
  __shared__ float sZ[4][BDIM][32];
  int wave = threadIdx.x >> 5, lane = threadIdx.x & 31;
  int n0 = blockIdx.x * 32;
  int n = lane & 15;
  int kk = (lane < 16) ? 0 : 2;
  int arow = (lane & 7) * HDIM;            // rows 8..15 duplicate 0..7 (results discarded)
  int col = wave * HDIM + n0 + n;          // column within 4H

  v8f acc0 = {}, acc1 = {};
  int aoff = arow + kk;
  int woff = kk * (4 * HDIM) + col;
#pragma unroll 8
  for (int k0 = 0; k0 < HDIM; k0 += 4) {
    v2f a = *(const v2f*)(wsx + aoff);
    v2f b0, b1;
    b0.x = Wx[woff];       b0.y = Wx[woff + 4 * HDIM];
    b1.x = Wx[woff + 16];  b1.y = Wx[woff + 4 * HDIM + 16];
    acc0 = wmma_f32(a, b0, acc0);
    acc1 = wmma_f32(a, b1, acc1);
    aoff += 4;
    woff += 16 * HDIM;
  }
  aoff = arow + kk;
  woff = kk * (4 * HDIM) + col;
#pragma unroll 8
  for (int k0 = 0; k0 < HDIM; k0 += 4) {
    v2f a = *(const v2f*)(wsh + aoff);
    v2f b0, b1;
    b0.x = Wh[woff];       b0.y = Wh[woff + 4 * HDIM];
    b1.x = Wh[woff + 16];  b1.y = Wh[woff + 4 * HDIM + 16];
    acc0 = wmma_f32(a, b0, acc0);
    acc1 = wmma_f32(a, b1, acc1);
    aoff += 4;
    woff += 16 * HDIM;
  }
  if (lane < 16) {
    float bb0 = bias[col], bb1 = bias[col + 16];
#pragma unroll
    for (int r = 0; r < 8; ++r) {
      sZ[wave][r][n] = acc0[r] + bb0;        // rows 0..7 = batch
      sZ[wave][r][n + 16] = acc1[r] + bb1;
    }
  }
  __syncthreads();
#pragma unroll
  for (int e = 0; e < 2; ++e) {
    int id = threadIdx.x + e * 128;          // 256 outputs, 128 threads
    int b = id >> 5, j = id & 31;
    float ig = sigmoidf_(sZ[0][b][j]);
    float fg = sigmoidf_(sZ[1][b][j]);
    float gg = tanhf(sZ[2][b][j]);
    float og = sigmoidf_(sZ[3][b][j]);
    int idx = b * HDIM + n0 + j;
    float cn = fg * wsc[idx] + ig * gg;
    wsc2[idx] = cn;
    wsh2[idx] = og * tanhf(cn);
  }
}

// ---------------- Per-step kernel 2: jt = tanh(enc_proj[b,t,:] + h2 @ W_pred) -------------
// grid = 16 blocks x 128 threads; one 16-col tile per wave (64 waves)
__global__ void pred_kernel(const float* __restrict__ wsh2, const float* __restrict__ W_pred,
                            const float* __restrict__ encp, float* __restrict__ wsjt,
                            int t, int T) {
  int wave = threadIdx.x >> 5, lane = threadIdx.x & 31;
  int n = lane & 15;
  int col = blockIdx.x * 64 + wave * 16 + n;
  int kk = (lane < 16) ? 0 : 2;
  int arow = (lane & 7) * HDIM;

  v8f acc0 = {};
  int aoff = arow + kk;
  int woff = kk * HDIM + col;
#pragma unroll 8
  for (int k0 = 0; k0 < HDIM; k0 += 4) {
    v2f a = *(const v2f*)(wsh2 + aoff);
    v2f b0;
    b0.x = W_pred[woff];
    b0.y = W_pred[woff + HDIM];
    acc0 = wmma_f32(a, b0, acc0);
    aoff += 4;
    woff += 4 * HDIM;
  }
  if (lane < 16) {
#pragma unroll
    for (int r = 0; r < 8; ++r) {
      int eb = (r * T + t) * HDIM + col;       // b_joint folded into encp
      wsjt[r * HDIM + col] = tanhf(acc0[r] + encp[eb]);
    }
  }
}

// ------- Per-step kernel 3: logits = jt @ W_out + b_out, fused argmax + update tail ------
__global__ void logits_kernel(const float* __restrict__ wsjt, const float* __restrict__ W_out,
                              const float* __restrict__ b_out,
                              unsigned long long* __restrict__ slots, int NV,
                              unsigned* __restrict__ doneCnt, int* __restrict__ last,
                              int* __restrict__ active, float* __restrict__ h,
                              float* __restrict__ c, const float* __restrict__ h2,
                              const float* __restrict__ c2, const int* __restrict__ lens,
                              const float* __restrict__ emb, float* __restrict__ wsx,
                              float* __restrict__ out, int t, int u, int T, int blank,
                              int isLast) {
  __shared__ unsigned long long sK[8][BDIM];
  __shared__ int sAdv[BDIM];
  __shared__ int sNl[BDIM];
  __shared__ int sLastBlk;
  int tid = threadIdx.x;
  int wave = tid >> 5, lane = tid & 31;
  int n = lane & 15;
  int c0 = blockIdx.x * 256 + wave * 32 + n;
  int c1 = c0 + 16;
  bool v0 = (lane < 16) && (c0 < NV);
  bool v1 = (lane < 16) && (c1 < NV);
  int c0c = c0 < NV ? c0 : NV - 1;             // clamp: dup cols masked at argmax
  int c1c = c1 < NV ? c1 : NV - 1;
  int kk = (lane < 16) ? 0 : 2;
  int arow = (lane & 7) * HDIM;

  v8f acc0 = {}, acc1 = {};
  int aoff = arow + kk;
  int w0 = kk * NV + c0c;
  int w1 = kk * NV + c1c;
#pragma unroll 8
  for (int k0 = 0; k0 < HDIM; k0 += 4) {
    v2f a = *(const v2f*)(wsjt + aoff);
    v2f b0, b1;
    b0.x = W_out[w0];  b0.y = W_out[w0 + NV];
    b1.x = W_out[w1];  b1.y = W_out[w1 + NV];
    acc0 = wmma_f32(a, b0, acc0);
    acc1 = wmma_f32(a, b1, acc1);
    aoff += 4;
    w0 += 4 * NV;
    w1 += 4 * NV;
  }
  float bo0 = b_out[c0c], bo1 = b_out[c1c];
  unsigned long long key[BDIM];
#pragma unroll
  for (int r = 0; r < BDIM; ++r) {
    unsigned long long ka = v0 ? packKey(acc0[r] + bo0, c0c) : 0ull;
    unsigned long long kb = v1 ? packKey(acc1[r] + bo1, c1c) : 0ull;
    key[r] = ka > kb ? ka : kb;
  }
#pragma unroll
  for (int off = 16; off >= 1; off >>= 1) {
#pragma unroll
    for (int r = 0; r < BDIM; ++r) {
      unsigned long long o = __shfl_xor(key[r], off);
      if (o > key[r]) key[r] = o;
    }
  }
  if (lane == 0) {
#pragma unroll
    for (int r = 0; r < BDIM; ++r) sK[wave][r] = key[r];
  }
  __syncthreads();
  if (tid < BDIM) {
    unsigned long long mx = 0ull;
#pragma unroll
    for (int w2 = 0; w2 < 8; ++w2) { unsigned long long v = sK[w2][tid]; if (v > mx) mx = v; }
    atomicMax(&slots[tid], mx);
  }
  // ---- arrival: last finished block runs the decode/commit tail ----
  __threadfence();          // this block's slot atomics globally visible before arrive
  __syncthreads();
  if (tid == 0) {
    unsigned prev = atomicAdd(doneCnt, 1u);
    sLastBlk = (prev == (unsigned)(gridDim.x - 1)) ? 1 : 0;
  }
  __syncthreads();
  if (!sLastBlk) return;

  if (tid == 0) *doneCnt = 0u;               // reset for next step (visible at kernel end)
  if (tid < BDIM) {
    unsigned long long keyv = atomicMax(&slots[tid], 0ull);   // coherent L2 read
    int sym = (int)(0xFFFFFFFFu - (unsigned)(keyv & 0xFFFFFFFFull));
    int act = (u == 0) ? (t < lens[tid] ? 1 : 0) : active[tid];
    int lab = act ? sym : blank;
    out[(size_t)tid * T * MAXSYM + (size_t)t * MAXSYM + u] = (float)lab;
    int a = (act && sym != blank) ? 1 : 0;
    int l2 = a ? sym : last[tid];
    last[tid] = l2;
    active[tid] = a;
    slots[tid] = 0ull;
    sAdv[tid] = a;
    sNl[tid] = l2;
  }
  __syncthreads();
  for (int i = tid; i < BDIM * HDIM; i += blockDim.x) {
    int b = i >> 10;
    if (sAdv[b]) { h[i] = h2[i]; c[i] = c2[i]; }
    wsx[i] = emb[(size_t)sNl[b] * HDIM + (i & 1023)];   // x for the next symbol step
  }
  if (isLast) {
    __syncthreads();
    size_t off = (size_t)BDIM * T * MAXSYM;
    for (int i = tid; i < BDIM * HDIM; i += blockDim.x) {
      out[off + i] = h[i];
      out[off + BDIM * HDIM + i] = c[i];
    }
  }
}

__global__ void init_kernel(float* h, float* c, float* wsx, int* last, int* active,
                            unsigned long long* slots, unsigned* doneCnt, int blank) {
  int tid = threadIdx.x;
  for (int i = tid; i < BDIM * HDIM; i += blockDim.x) { h[i] = 0.f; c[i] = 0.f; wsx[i] = 0.f; }
  if (tid < BDIM) { last[tid] = blank; active[tid] = 0; slots[tid] = 0ull; }
  if (tid == 0) *doneCnt = 0u;
}

extern "C" void kernel_launch(void* const* d_in, const int* in_sizes, int n_in,
                              void* d_out, int out_size, void* d_ws, size_t ws_size,
                              hipStream_t stream) {
  const float* enc     = (const float*)d_in[0];
  const int*   lens    = (const int*)  d_in[1];
  const float* emb     = (const float*)d_in[2];
  const float* Wx      = (const float*)d_in[3];
  const float* Wh      = (const float*)d_in[4];
  const float* bias    = (const float*)d_in[5];
  const float* W_enc   = (const float*)d_in[6];
  const float* W_pred  = (const float*)d_in[7];
  const float* b_joint = (const float*)d_in[8];
  const float* W_out   = (const float*)d_in[9];
  const float* b_out   = (const float*)d_in[10];

  int B  = in_sizes[1];                    // 8
  int T  = in_sizes[0] / (B * HDIM);       // 200
  int NV = in_sizes[2] / HDIM;             // V+1 = 16385
  int blank = NV - 1;
  int M  = B * T;
  (void)n_in; (void)out_size; (void)ws_size;

  float* ws   = (float*)d_ws;
  float* encp = ws;
  size_t off = (size_t)M * HDIM;
  float* h   = ws + off; off += (size_t)B * HDIM;
  float* cst = ws + off; off += (size_t)B * HDIM;
  float* h2  = ws + off; off += (size_t)B * HDIM;
  float* c2  = ws + off; off += (size_t)B * HDIM;
  float* jt  = ws + off; off += (size_t)B * HDIM;
  float* wsx = ws + off; off += (size_t)B * HDIM;
  off = (off + 1) & ~(size_t)1;            // 8B align for u64 slots
  unsigned long long* slots = (unsigned long long*)(ws + off); off += 2 * B;
  int* last   = (int*)(ws + off); off += B;
  int* active = (int*)(ws + off); off += B;
  unsigned* doneCnt = (unsigned*)(ws + off); off += 1;

  init_kernel<<<1, 256, 0, stream>>>(h, cst, wsx, last, active, slots, doneCnt, blank);

  dim3 egrid(HDIM / 128, (M + 15) / 16);
  enc_proj_kernel<<<egrid, 128, 0, stream>>>(enc, W_enc, b_joint, encp, T, M);

  float* out = (float*)d_out;
  int nLogitBlocks = (NV + 255) / 256;
  for (int t = 0; t < T; ++t) {
    for (int u = 0; u < MAXSYM; ++u) {
      lstm_kernel<<<HDIM / 32, 128, 0, stream>>>(wsx, h, Wx, Wh, bias, cst, h2, c2);
      pred_kernel<<<HDIM / 64, 128, 0, stream>>>(h2, W_pred, encp, jt, t, T);
      int isLast = (t == T - 1 && u == MAXSYM - 1) ? 1 : 0;
      logits_kernel<<<nLogitBlocks, 256, 0, stream>>>(
          jt, W_out, b_out, slots, NV, doneCnt, last, active, h, cst, h2, c2,
          lens, emb, wsx, out, t, u, T, blank, isLast);
    }
  }
}